// VQ_90512140796326
// MI455X (gfx1250) — compile-verified
//
#include <hip/hip_runtime.h>

// ---------------------------------------------------------------------------
// VQ-VAE vector quantization for MI455X (gfx1250), fp32 WMMA path.
//   x:   [32, 64, 64, 64] fp32 NCHW  -> vectors x[b,:,h,w], elements 4096 apart
//   emb: [512, 64] fp32 codebook
//   out: [0] = c_loss, [1..8388608] = quantized, NCHW
// GEMM core: V_WMMA_F32_16X16X4_F32, K=64 -> 16 chained WMMAs per 16x16 tile.
// ---------------------------------------------------------------------------

typedef float v2f __attribute__((ext_vector_type(2)));
typedef float v8f __attribute__((ext_vector_type(8)));

#define NUM_EMB    512
#define DIM        64
#define HW         4096        // 64*64 spatial positions per image
#define ROW_TILE   16
#define WAVES_WG   8
#define THREADS    256
#define N_BLOCKS   1024        // 131072 rows / (16 rows * 8 waves)
#define EMB_STRIDE 66          // padded LDS row stride (words) -> no bank conflicts
#define SMEM_WORDS (NUM_EMB * EMB_STRIDE + NUM_EMB + WAVES_WG * ROW_TILE)
#define SMEM_BYTES (SMEM_WORDS * 4)

__global__ __launch_bounds__(THREADS)
void vq_argmin_quantize(const float* __restrict__ X,
                        const float* __restrict__ E,
                        float* __restrict__ out) {
  extern __shared__ float smem[];
  float* emb_s  = smem;                                  // 512 x 66
  float* norm_s = smem + NUM_EMB * EMB_STRIDE;           // 512
  int*   idx_s  = (int*)(norm_s + NUM_EMB);              // 8 waves x 16 rows

  const int tid  = threadIdx.x;
  const int wave = tid >> 5;
  const int lane = tid & 31;
  const int m    = lane & 15;   // row-in-tile (A) / code-in-tile (B,C)
  const int kh   = lane >> 4;   // K-half selector per ISA 32-bit layouts

  // ---- stage codebook into padded LDS (float2 granularity, coalesced) ----
  const float2* Eg = (const float2*)E;
  #pragma unroll 4
  for (int i = 0; i < 64; ++i) {            // 16384 float2 / 256 threads
    int p   = tid + THREADS * i;
    int row = p >> 5;                       // 32 float2 per codebook row
    int j   = p & 31;
    float2 v = Eg[row * 32 + j];
    *(float2*)&emb_s[row * EMB_STRIDE + 2 * j] = v;
  }
  __syncthreads();

  // ---- code norms ||e_k||^2 ----
  for (int k = tid; k < NUM_EMB; k += THREADS) {
    const float* r = &emb_s[k * EMB_STRIDE];
    float s = 0.f;
    #pragma unroll
    for (int d = 0; d < DIM; ++d) s = fmaf(r[d], r[d], s);
    norm_s[k] = s;
  }
  __syncthreads();

  // ---- this wave's 16-row tile ----
  const int  tile = blockIdx.x * WAVES_WG + wave;        // 0..8191
  const int  row0 = tile * ROW_TILE;
  const int  b    = row0 >> 12;                          // image index
  const int  hw0  = row0 & 4095;                         // spatial base
  const long bofs = (long)b * (DIM * HW);

  // ---- preload A fragments: afrag[s] holds x[row0+m, 4s+2kh .. +1] ----
  // (ISA 16x4 f32 A layout: lanes 0-15/16-31 -> M=0..15, VGPR0/1 -> K even/odd)
  v2f afrag[16];
  #pragma unroll
  for (int s = 0; s < 16; ++s) {
    int d = 4 * s + 2 * kh;
    afrag[s].x = X[bofs + (long)d       * HW + hw0 + m];
    afrag[s].y = X[bofs + (long)(d + 1) * HW + hw0 + m];
  }

  float bestv[8];
  int   besti[8];
  #pragma unroll
  for (int r = 0; r < 8; ++r) { bestv[r] = 3.402823e38f; besti[r] = 0; }

  // ---- sweep 32 code tiles: Gram via chained f32 WMMA, then running argmin ----
  for (int cb = 0; cb < NUM_EMB; cb += 16) {
    v8f c = {};
    #pragma unroll
    for (int s = 0; s < 16; ++s) {
      int d = 4 * s + 2 * kh;
      v2f bf = *(const v2f*)&emb_s[(cb + m) * EMB_STRIDE + d];  // ds_load_b64
      c = __builtin_amdgcn_wmma_f32_16x16x4_f32(
              false, afrag[s], false, bf, (short)0, c, false, false);
    }
    // C layout: lane n = m (code), VGPR r -> row r + 8*kh
    float nrm = norm_s[cb + m];
    #pragma unroll
    for (int r = 0; r < 8; ++r) {
      float d2 = fmaf(-2.f, c[r], nrm);    // ||x||^2 dropped (argmin-invariant)
      if (d2 < bestv[r]) { bestv[r] = d2; besti[r] = cb + m; }
    }
  }

  // ---- reduce argmin across the 16 code lanes (per half-wave) ----
  #pragma unroll
  for (int r = 0; r < 8; ++r) {
    float v = bestv[r];
    int   i = besti[r];
    #pragma unroll
    for (int s = 1; s < 16; s <<= 1) {
      float ov = __shfl_xor(v, s, 16);
      int   oi = __shfl_xor(i, s, 16);
      if (ov < v || (ov == v && oi < i)) { v = ov; i = oi; }  // lowest-index ties
    }
    besti[r] = i;
  }

  // lanes 0 / 16 publish row indices (rows 0-7 / 8-15) for per-lane gather
  if (m == 0) {
    #pragma unroll
    for (int r = 0; r < 8; ++r)
      idx_s[wave * ROW_TILE + (kh ? 8 + r : r)] = besti[r];
  }
  __syncthreads();

  // ---- gather + straight-through output + loss partial ----
  const int    myidx = idx_s[wave * ROW_TILE + m];
  const float* qrow  = &emb_s[myidx * EMB_STRIDE];
  float acc = 0.f;
  #pragma unroll 4
  for (int it = 0; it < 32; ++it) {
    int   dd   = 2 * it + kh;
    long  off  = bofs + (long)dd * HW + hw0 + m;    // lanes contiguous in m
    float q    = qrow[dd];
    float xv   = X[off];
    float diff = q - xv;
    out[1 + off] = xv + diff;                       // inp + (quantized - inp)
    acc = fmaf(diff, diff, acc);
  }

  // ---- loss reduction: c_loss = 1.25 * mean(diff^2) ----
  #pragma unroll
  for (int s = 1; s < 32; s <<= 1) acc += __shfl_xor(acc, s, 32);
  if (lane == 0) atomicAdd(out, acc * (1.25f / 8388608.f));
}

extern "C" void kernel_launch(void* const* d_in, const int* in_sizes, int n_in,
                              void* d_out, int out_size, void* d_ws, size_t ws_size,
                              hipStream_t stream) {
  (void)in_sizes; (void)n_in; (void)d_ws; (void)ws_size; (void)out_size;
  const float* X = (const float*)d_in[0];
  const float* E = (const float*)d_in[1];
  float* out = (float*)d_out;

  // allow >64KB dynamic LDS (137,728 B -> 2 workgroups per 320KB WGP)
  static_assert(SMEM_BYTES < 320 * 1024, "LDS over WGP budget");
  hipFuncSetAttribute((const void*)vq_argmin_quantize,
                      hipFuncAttributeMaxDynamicSharedMemorySize, SMEM_BYTES);

  hipMemsetAsync(out, 0, sizeof(float), stream);  // loss accumulator
  vq_argmin_quantize<<<N_BLOCKS, THREADS, SMEM_BYTES, stream>>>(X, E, out);
}